// Model_47588237639752
// MI455X (gfx1250) — compile-verified
//
#include <hip/hip_runtime.h>
#include <hip/hip_bf16.h>
#include <stdint.h>

typedef __attribute__((ext_vector_type(2))) float v2f;
typedef __attribute__((ext_vector_type(8))) float v8f;
typedef __attribute__((ext_vector_type(4))) unsigned int u32x4;
typedef __attribute__((ext_vector_type(8))) int i32x8;
typedef __attribute__((ext_vector_type(4))) int i32x4;

#define N_ROWS 2048
#define D_IN   1024
#define H_DIM  1024
#define D_OUTC 512

// GEMM1 block tiling
#define BM1 128
#define BN1 64
#define KC  32
#define LDSROW 34   // 32 data DWORDs + 2 pad DWORDs (TDM pad_enable) per row

static __device__ __forceinline__ v8f wmma_f32(v2f a, v2f b, v8f c) {
  // V_WMMA_F32_16X16X4_F32 : D = A(16x4) * B(4x16) + C(16x16), all fp32
  return __builtin_amdgcn_wmma_f32_16x16x4_f32(false, a, false, b, (short)0, c,
                                               false, false);
}

#if __has_builtin(__builtin_amdgcn_tensor_load_to_lds) && \
    __has_builtin(__builtin_amdgcn_s_wait_tensorcnt)
#define HAVE_TDM 1
#endif

#ifdef HAVE_TDM
// Issue a TDM 2D tile load: tile = KC x BM1 fp32 from a row-major tensor
// (width tensor_w, height tensor_h, row stride tensor_w), starting at gptr,
// into LDS at byte offset lds_off, with 2 pad DWORDs every 32 DWORDs.
static __device__ __forceinline__ void tdm_stage_a(const float* gptr,
                                                   uint32_t lds_off) {
  const uint64_t ga = (uint64_t)(uintptr_t)gptr;
  u32x4 g0;
  g0.x = 1u;                                    // count=1 (valid user D#)
  g0.y = lds_off;                               // LDS byte address
  g0.z = (uint32_t)(ga & 0xFFFFFFFFu);          // global_addr[31:0]
  g0.w = (uint32_t)((ga >> 32) & 0x1FFFFFFu)    // global_addr[56:32]
       | (2u << 30);                            // type=2 ("image")
  i32x8 g1;
  g1[0] = (2 << 16)                             // data_size = 4 bytes
        | (1 << 20)                             // pad_enable
        | (4 << 22)                             // pad_interval: 32 DWORDs
        | (1 << 25);                            // pad_amount:   2 DWORDs
  g1[1] = (D_IN & 0xFFFF) << 16;                // tensor_dim0[15:0]
  g1[2] = ((D_IN >> 16) & 0xFFFF)               // tensor_dim0[31:16]
        | ((N_ROWS & 0xFFFF) << 16);            // tensor_dim1[15:0]
  g1[3] = ((N_ROWS >> 16) & 0xFFFF)             // tensor_dim1[31:16]
        | (KC << 16);                           // tile_dim0 = KC
  g1[4] = BM1;                                  // tile_dim1 = 128 rows
  g1[5] = D_IN;                                 // tensor_dim0_stride[31:0]
  g1[6] = 0;                                    // strides hi / dim1_stride lo
  g1[7] = 0;
  i32x4 z4 = {0, 0, 0, 0};
#if defined(__clang_major__) && __clang_major__ >= 23
  i32x8 z8 = {0, 0, 0, 0, 0, 0, 0, 0};
  __builtin_amdgcn_tensor_load_to_lds(g0, g1, z4, z4, z8, 0);
#else
  __builtin_amdgcn_tensor_load_to_lds(g0, g1, z4, z4, 0);
#endif
}
#endif

// ---------------- deterministic PRNG + Gamma sampling ----------------
static __device__ __forceinline__ uint32_t pcg_next(uint32_t& s) {
  s = s * 747796405u + 2891336453u;
  uint32_t w = ((s >> ((s >> 28u) + 4u)) ^ s) * 277803737u;
  return (w >> 22u) ^ w;
}
static __device__ __forceinline__ float unif01(uint32_t& s) {
  return ((float)(pcg_next(s) >> 8) + 0.5f) * (1.0f / 16777216.0f);
}
static __device__ __forceinline__ float softplusf(float x) {
  return (x > 20.0f) ? x : log1pf(__expf(x));
}
// Marsaglia-Tsang Gamma(alpha, 1) sampler (rsample-equivalent, deterministic)
static __device__ float gamma_sample(float alpha, uint32_t seed) {
  uint32_t s = seed;
  float boost = 1.0f;
  if (alpha < 1.0f) {
    float u = unif01(s);
    boost = __powf(u, 1.0f / alpha);
    alpha += 1.0f;
  }
  const float d = alpha - (1.0f / 3.0f);
  const float c = rsqrtf(9.0f * d);
  for (int it = 0; it < 64; ++it) {
    float u1 = unif01(s), u2 = unif01(s);
    float x = sqrtf(-2.0f * __logf(u1)) * __cosf(6.28318530718f * u2);
    float v = 1.0f + c * x;
    if (v <= 0.0f) continue;
    v = v * v * v;
    float u = unif01(s);
    if (__logf(u) < 0.5f * x * x + d - d * v + d * __logf(v))
      return d * v * boost;
  }
  return d * boost;
}

// S[-1]: one Gamma vector for sample index 99, normalized (Dirichlet)
__global__ void __launch_bounds__(1024)
dirichlet_last_kernel(const float* __restrict__ parameter,
                      float* __restrict__ S_ws,
                      float* __restrict__ out_tail) {
  __shared__ float red[1024];
  const int i = threadIdx.x;
  const float phi = softplusf(parameter[i]);
  uint32_t seed = 0x9E3779B9u ^ (99u * 0x85EBCA6Bu) ^ ((uint32_t)i * 0xC2B2AE35u);
  const float g = gamma_sample(phi, seed);
  red[i] = g;
  __syncthreads();
  for (int off = 512; off > 0; off >>= 1) {
    if (i < off) red[i] += red[i + off];
    __syncthreads();
  }
  const float sval = g / red[0];
  S_ws[i] = sval;
  out_tail[i] = sval;   // second tuple output: S[-1]
}

// ---------------- GEMM1: h = relu((x@W1 - b1) * S) ----------------
// A tile (128 x 32) staged into LDS by the Tensor Data Mover, double-buffered.
// Wave tile 32x32 (2x2 WMMA), 8 waves -> block tile 128(M) x 64(N).
__global__ void __launch_bounds__(256)
gemm1_scale_relu(const float* __restrict__ A,    // x   [2048,1024]
                 const float* __restrict__ B,    // W1  [1024,1024]
                 const float* __restrict__ bias, // b1  [1024]
                 const float* __restrict__ S,    // [1024]
                 float* __restrict__ Hout) {     // h   [2048,1024]
  __shared__ float ldsA[2 * BM1 * LDSROW];       // ~34 KB, double-buffered
  const int tid  = threadIdx.x;
  const int lane = tid & 31;
  const int wave = tid >> 5;
  const int wm = wave & 3, wn = wave >> 2;
  const int m_blk = blockIdx.y * BM1;
  const int m0 = m_blk + wm * 32;
  const int n0 = blockIdx.x * BN1 + wn * 32;
  const int row = lane & 15;
  const int kb  = (lane >> 4) << 1;   // 0 or 2
  const int col = lane & 15;

  v8f acc00 = {}, acc01 = {}, acc10 = {}, acc11 = {};

#ifdef HAVE_TDM
  const uint32_t lds_base = (uint32_t)(uintptr_t)&ldsA[0];
  // prologue: stage chunk 0 into buffer 0
  if (wave == 0) {
    tdm_stage_a(A + (size_t)m_blk * D_IN, lds_base);
    __builtin_amdgcn_s_wait_tensorcnt(0);
  }
  __syncthreads();
#else
  // fallback: cooperative staged copy (same LDS layout incl. padding)
  {
    const float* gp = A + (size_t)m_blk * D_IN;
    for (int idx = tid; idx < BM1 * KC; idx += 256) {
      int r = idx >> 5, c = idx & (KC - 1);
      ldsA[r * LDSROW + c] = gp[(size_t)r * D_IN + c];
    }
  }
  __syncthreads();
#endif

  int buf = 0;
  for (int k0 = 0; k0 < D_IN; k0 += KC) {
    const int nbuf = buf ^ 1;
    // stage next chunk while computing on the current one
    if (k0 + KC < D_IN) {
#ifdef HAVE_TDM
      if (wave == 0)
        tdm_stage_a(A + (size_t)m_blk * D_IN + (k0 + KC),
                    lds_base + (uint32_t)(nbuf * BM1 * LDSROW * 4));
#else
      const float* gp = A + (size_t)m_blk * D_IN + (k0 + KC);
      for (int idx = tid; idx < BM1 * KC; idx += 256) {
        int r = idx >> 5, c = idx & (KC - 1);
        ldsA[nbuf * BM1 * LDSROW + r * LDSROW + c] = gp[(size_t)r * D_IN + c];
      }
#endif
    }

    const float* la0 = &ldsA[buf * BM1 * LDSROW + (wm * 32 + row) * LDSROW + kb];
    const float* la1 = la0 + 16 * LDSROW;
#pragma unroll
    for (int kk = 0; kk < KC; kk += 4) {
      v2f a0 = *(const v2f*)(la0 + kk);          // ds_load_b64, 8B-aligned
      v2f a1 = *(const v2f*)(la1 + kk);
      const float* bp = B + (size_t)(k0 + kk + kb) * H_DIM;
      v2f b0, b1v;
      b0.x  = bp[n0 + col];          b0.y  = bp[H_DIM + n0 + col];
      b1v.x = bp[n0 + 16 + col];     b1v.y = bp[H_DIM + n0 + 16 + col];
      acc00 = wmma_f32(a0, b0,  acc00);
      acc01 = wmma_f32(a0, b1v, acc01);
      acc10 = wmma_f32(a1, b0,  acc10);
      acc11 = wmma_f32(a1, b1v, acc11);
    }

#ifdef HAVE_TDM
    if (wave == 0 && k0 + KC < D_IN) __builtin_amdgcn_s_wait_tensorcnt(0);
#endif
    __syncthreads();
    buf = nbuf;
  }

  const int rbase = (lane < 16) ? 0 : 8;
  const int c0 = n0 + col, c1 = n0 + 16 + col;
  const float s0 = S[c0], s1 = S[c1];
  const float bb0 = bias[c0], bb1 = bias[c1];
#pragma unroll
  for (int v = 0; v < 8; ++v) {
    const int r0 = m0 + v + rbase;
    const int r1 = m0 + 16 + v + rbase;
    float p;
    p = (acc00[v] - bb0) * s0; Hout[(size_t)r0 * H_DIM + c0] = p > 0.0f ? p : 0.0f;
    p = (acc01[v] - bb1) * s1; Hout[(size_t)r0 * H_DIM + c1] = p > 0.0f ? p : 0.0f;
    p = (acc10[v] - bb0) * s0; Hout[(size_t)r1 * H_DIM + c0] = p > 0.0f ? p : 0.0f;
    p = (acc11[v] - bb1) * s1; Hout[(size_t)r1 * H_DIM + c1] = p > 0.0f ? p : 0.0f;
  }
}

// ---------------- GEMM2: out = sigmoid(h@W2 + b2) ----------------
__global__ void __launch_bounds__(256)
gemm2_sigmoid(const float* __restrict__ A,    // h   [2048,1024]
              const float* __restrict__ B,    // W2  [1024,512]
              const float* __restrict__ bias, // b2  [512]
              float* __restrict__ Out) {      // [2048,512]
  const int lane = threadIdx.x & 31;
  const int wave = threadIdx.x >> 5;
  const int wm = wave & 3, wn = wave >> 2;
  const int m0 = blockIdx.y * 128 + wm * 32;
  const int n0 = blockIdx.x * 64 + wn * 32;
  const int row = lane & 15;
  const int kb  = (lane >> 4) << 1;
  const int col = lane & 15;

  v8f acc00 = {}, acc01 = {}, acc10 = {}, acc11 = {};
  const float* a0p = A + (size_t)(m0 + row) * H_DIM + kb;
  const float* a1p = A + (size_t)(m0 + 16 + row) * H_DIM + kb;
  for (int k = 0; k < H_DIM; k += 4) {
    v2f a0 = *(const v2f*)(a0p + k);
    v2f a1 = *(const v2f*)(a1p + k);
    const float* bp = B + (size_t)(k + kb) * D_OUTC;
    v2f b0, b1v;
    b0.x  = bp[n0 + col];            b0.y  = bp[D_OUTC + n0 + col];
    b1v.x = bp[n0 + 16 + col];       b1v.y = bp[D_OUTC + n0 + 16 + col];
    acc00 = wmma_f32(a0, b0,  acc00);
    acc01 = wmma_f32(a0, b1v, acc01);
    acc10 = wmma_f32(a1, b0,  acc10);
    acc11 = wmma_f32(a1, b1v, acc11);
  }
  const int rbase = (lane < 16) ? 0 : 8;
  const int c0 = n0 + col, c1 = n0 + 16 + col;
  const float bb0 = bias[c0], bb1 = bias[c1];
#pragma unroll
  for (int v = 0; v < 8; ++v) {
    const int r0 = m0 + v + rbase;
    const int r1 = m0 + 16 + v + rbase;
    float p;
    p = acc00[v] + bb0; Out[(size_t)r0 * D_OUTC + c0] = 1.0f / (1.0f + __expf(-p));
    p = acc01[v] + bb1; Out[(size_t)r0 * D_OUTC + c1] = 1.0f / (1.0f + __expf(-p));
    p = acc10[v] + bb0; Out[(size_t)r1 * D_OUTC + c0] = 1.0f / (1.0f + __expf(-p));
    p = acc11[v] + bb1; Out[(size_t)r1 * D_OUTC + c1] = 1.0f / (1.0f + __expf(-p));
  }
}

extern "C" void kernel_launch(void* const* d_in, const int* in_sizes, int n_in,
                              void* d_out, int out_size, void* d_ws, size_t ws_size,
                              hipStream_t stream) {
  const float* x  = (const float*)d_in[0];
  const float* W1 = (const float*)d_in[1];
  const float* b1 = (const float*)d_in[2];
  const float* W2 = (const float*)d_in[3];
  const float* b2 = (const float*)d_in[4];
  const float* parameter = (const float*)d_in[5];

  float* out  = (float*)d_out;                 // labels[-1]: 2048*512, then S[-1]: 1024
  float* S_ws = (float*)d_ws;                  // 1024 floats
  float* Hbuf = (float*)d_ws + 1024;           // 2048*1024 floats (8 MB)

  dirichlet_last_kernel<<<1, 1024, 0, stream>>>(
      parameter, S_ws, out + (size_t)N_ROWS * D_OUTC);

  gemm1_scale_relu<<<dim3(H_DIM / BN1, N_ROWS / BM1), 256, 0, stream>>>(
      x, W1, b1, S_ws, Hbuf);

  gemm2_sigmoid<<<dim3(D_OUTC / 64, N_ROWS / 128), 256, 0, stream>>>(
      Hbuf, W2, b2, out);
}